// QuantumInspiredSemanticSpaceV4_85950885528251
// MI455X (gfx1250) — compile-verified
//
#include <hip/hip_runtime.h>
#include <math.h>

#define B_SZ 1024
#define D_SZ 2048
#define M_SZ 8192
#define Q_SZ 4
#define MOM  0.95f
#define TWO_PI 6.283185307179586f

typedef __attribute__((ext_vector_type(2))) float v2f;
typedef __attribute__((ext_vector_type(8))) float v8f;

// ---------------- mean (over Q) + L2-normalize rows ----------------
__global__ __launch_bounds__(256)
void k_mean_norm(const float* __restrict__ src, float* __restrict__ dst, int qcount) {
  const int row = blockIdx.x;
  const int tid = threadIdx.x;
  __shared__ float red[256];
  const size_t sbase = (size_t)row * qcount * D_SZ;
  const size_t dbase = (size_t)row * D_SZ;
  const float qinv = 1.0f / (float)qcount;
  float ss = 0.f;
  for (int d = tid; d < D_SZ; d += 256) {
    float v = 0.f;
    for (int q = 0; q < qcount; ++q) v += src[sbase + (size_t)q * D_SZ + d];
    v *= qinv;
    dst[dbase + d] = v;
    ss += v * v;
  }
  red[tid] = ss;
  __syncthreads();
  for (int s = 128; s > 0; s >>= 1) { if (tid < s) red[tid] += red[tid + s]; __syncthreads(); }
  const float inv = 1.0f / fmaxf(sqrtf(red[0]), 1e-8f);
  for (int d = tid; d < D_SZ; d += 256) dst[dbase + d] *= inv;
}

// ---------------- sims = pn @ mn^T via V_WMMA_F32_16X16X4_F32 ----------------
// grid: x = N/256 (=32), y = B/16 (=64); block = 128 (4 waves), wave -> 16x64 strip
__global__ __launch_bounds__(128)
void k_gemm_wmma(const float* __restrict__ pn, const float* __restrict__ mn,
                 float* __restrict__ sims) {
  const int wave = threadIdx.x >> 5;
  const int lane = threadIdx.x & 31;
  const int row0 = blockIdx.y * 16;
  const int col0 = blockIdx.x * 256 + wave * 64;
  const int idx16 = lane & 15;       // m for A, n for B
  const int khalf = lane >> 4;       // 0 or 1 -> K = 2*khalf, 2*khalf+1

  const float* aptr  = pn + (size_t)(row0 + idx16) * D_SZ + 2 * khalf;
  const float* bptr  = mn + (size_t)(col0 + idx16) * D_SZ + 2 * khalf;

  v8f acc0 = {}, acc1 = {}, acc2 = {}, acc3 = {};
  #pragma unroll 4
  for (int k = 0; k < D_SZ; k += 4) {
    v2f a  = *(const v2f*)(aptr + k);
    v2f b0 = *(const v2f*)(bptr + k);
    v2f b1 = *(const v2f*)(bptr + 16 * (size_t)D_SZ + k);
    v2f b2 = *(const v2f*)(bptr + 32 * (size_t)D_SZ + k);
    v2f b3 = *(const v2f*)(bptr + 48 * (size_t)D_SZ + k);
    acc0 = __builtin_amdgcn_wmma_f32_16x16x4_f32(false, a, false, b0, (short)0, acc0, false, false);
    acc1 = __builtin_amdgcn_wmma_f32_16x16x4_f32(false, a, false, b1, (short)0, acc1, false, false);
    acc2 = __builtin_amdgcn_wmma_f32_16x16x4_f32(false, a, false, b2, (short)0, acc2, false, false);
    acc3 = __builtin_amdgcn_wmma_f32_16x16x4_f32(false, a, false, b3, (short)0, acc3, false, false);
  }

  const int n = lane & 15;
  const int mbase = (lane >> 4) * 8;
  #pragma unroll
  for (int v = 0; v < 8; ++v) {
    size_t r = (size_t)(row0 + mbase + v) * M_SZ + col0;
    sims[r +  0 + n] = acc0[v];
    sims[r + 16 + n] = acc1[v];
    sims[r + 32 + n] = acc2[v];
    sims[r + 48 + n] = acc3[v];
  }
}

// ---------------- per-row sum/sumsq/max/argmax ----------------
__global__ __launch_bounds__(256)
void k_rowstats(const float* __restrict__ sims, float* __restrict__ rowsum,
                float* __restrict__ rowsq, float* __restrict__ bestsim,
                int* __restrict__ bestidx) {
  const int b = blockIdx.x, tid = threadIdx.x;
  __shared__ float smax[256]; __shared__ int sidx[256];
  __shared__ float ssum[256]; __shared__ float ssq[256];
  const float* row = sims + (size_t)b * M_SZ;
  float mx = -INFINITY; int mi = 0; float s = 0.f, sq = 0.f;
  for (int d = tid; d < M_SZ; d += 256) {
    float v = row[d];
    s += v; sq += v * v;
    if (v > mx) { mx = v; mi = d; }
  }
  smax[tid] = mx; sidx[tid] = mi; ssum[tid] = s; ssq[tid] = sq;
  __syncthreads();
  for (int st = 128; st > 0; st >>= 1) {
    if (tid < st) {
      ssum[tid] += ssum[tid + st]; ssq[tid] += ssq[tid + st];
      float ov = smax[tid + st]; int oi = sidx[tid + st];
      if (ov > smax[tid] || (ov == smax[tid] && oi < sidx[tid])) { smax[tid] = ov; sidx[tid] = oi; }
    }
    __syncthreads();
  }
  if (tid == 0) { rowsum[b] = ssum[0]; rowsq[b] = ssq[0]; bestsim[b] = smax[0]; bestidx[b] = sidx[0]; }
}

// ---------------- global stats + MLP threshold ----------------
__global__ __launch_bounds__(256)
void k_thresh(const float* __restrict__ rowsum, const float* __restrict__ rowsq,
              const float* __restrict__ bestsim, const float* __restrict__ usage,
              const float* __restrict__ W1, const float* __restrict__ b1,
              const float* __restrict__ W2, const float* __restrict__ b2,
              float* __restrict__ thresh_ws, float* __restrict__ thresh_out) {
  const int tid = threadIdx.x;
  __shared__ float r0[256], r1[256], r2[256], r3[256];
  float s = 0.f, sq = 0.f, mx = -INFINITY, us = 0.f;
  for (int i = tid; i < B_SZ; i += 256) { s += rowsum[i]; sq += rowsq[i]; mx = fmaxf(mx, bestsim[i]); }
  for (int i = tid; i < M_SZ; i += 256) us += usage[i];
  r0[tid] = s; r1[tid] = sq; r2[tid] = mx; r3[tid] = us;
  __syncthreads();
  for (int st = 128; st > 0; st >>= 1) {
    if (tid < st) {
      r0[tid] += r0[tid + st]; r1[tid] += r1[tid + st];
      r2[tid] = fmaxf(r2[tid], r2[tid + st]); r3[tid] += r3[tid + st];
    }
    __syncthreads();
  }
  __shared__ float feats[20];
  __shared__ float h[64];
  if (tid < 20) feats[tid] = 0.f;
  __syncthreads();
  if (tid == 0) {
    const float N = (float)B_SZ * (float)M_SZ;
    float mean = r0[0] / N;
    float var = (r1[0] - N * mean * mean) / (N - 1.0f);
    feats[0] = mean;
    feats[1] = sqrtf(fmaxf(var, 0.f));
    feats[2] = r2[0];
    feats[3] = 1.0f;
    feats[4] = r3[0] / (float)M_SZ;
    feats[6] = 0.5f;
    feats[7] = 0.1f;
  }
  __syncthreads();
  if (tid < 64) {
    float a = b1[tid];
    for (int k = 0; k < 20; ++k) a += feats[k] * W1[tid * 20 + k];
    h[tid] = 0.5f * a * (1.0f + erff(a * 0.70710678118654752f));   // exact GELU
  }
  __syncthreads();
  if (tid == 0) {
    float o = b2[0];
    for (int j = 0; j < 64; ++j) o += h[j] * W2[j];
    float t = 1.0f / (1.0f + expf(-o));
    thresh_ws[0] = t;
    thresh_out[0] = t;
  }
}

// ---------------- importance + clear aux arrays ----------------
__global__ __launch_bounds__(256)
void k_imp(const float* __restrict__ usage, const int* __restrict__ last_access,
           const int* __restrict__ time_step, float* __restrict__ imp,
           int* __restrict__ owner, int* __restrict__ matchcnt) {
  int m = blockIdx.x * blockDim.x + threadIdx.x;
  if (m < M_SZ) {
    imp[m] = usage[m] * expf(-0.001f * (float)(time_step[0] - last_access[m]));
    owner[m] = -1;
    matchcnt[m] = 0;
  }
}

// ---------------- exact stable top-k (ascending importance) via rank count ----------------
__global__ __launch_bounds__(256)
void k_pool(const float* __restrict__ imp, int* __restrict__ pool) {
  int m = blockIdx.x * blockDim.x + threadIdx.x;
  if (m >= M_SZ) return;
  float im = imp[m];
  int r = 0;
  for (int j = 0; j < M_SZ; ++j) {
    float v = imp[j];
    r += (v < im) || (v == im && j < m);
  }
  if (r < B_SZ) pool[r] = m;
}

// ---------------- novelty, rank scan, meaning ids, owner/matchcnt ----------------
__global__ __launch_bounds__(1024)
void k_novel(const float* __restrict__ bestsim, const int* __restrict__ bestidx,
             const float* __restrict__ thresh, const int* __restrict__ pool,
             int* __restrict__ novel_ws, int* __restrict__ mids_ws,
             int* __restrict__ owner, int* __restrict__ matchcnt,
             float* __restrict__ out_meaning, float* __restrict__ out_isnovel,
             float* __restrict__ out_conf) {
  const int b = threadIdx.x;
  __shared__ int nv[B_SZ];
  __shared__ int cum[B_SZ];
  const int isn = (bestsim[b] < thresh[0]) ? 1 : 0;
  nv[b] = isn;
  __syncthreads();
  if (b == 0) { int s = 0; for (int j = 0; j < B_SZ; ++j) { s += nv[j]; cum[j] = s; } }
  __syncthreads();
  int rank = cum[b] - 1;
  rank = rank < 0 ? 0 : (rank > B_SZ - 1 ? B_SZ - 1 : rank);
  const int rep = pool[rank];
  const int bi  = bestidx[b];
  const int mid = isn ? rep : bi;
  novel_ws[b] = isn;
  mids_ws[b]  = mid;
  out_meaning[b] = (float)mid;
  out_isnovel[b] = (float)isn;
  out_conf[b]    = 1.0f - (float)isn;
  if (isn) owner[rep] = b;                 // ranks unique -> no race
  else atomicAdd(&matchcnt[bi], 1);
}

// ---------------- memory update: sequential momentum collapse per slot + novel overwrite ----------------
__global__ __launch_bounds__(256)
void k_mem(const float* __restrict__ qm, const float* __restrict__ qph,
           const float* __restrict__ patterns, const float* __restrict__ rand_scale,
           const float* __restrict__ rand_phase, const int* __restrict__ novel_ws,
           const int* __restrict__ bestidx_ws, const int* __restrict__ owner,
           const int* __restrict__ matchcnt, float* __restrict__ mem_out,
           float* __restrict__ ph_out) {
  const int i = blockIdx.x;
  const int tid = threadIdx.x;
  __shared__ float acc[D_SZ];
  const int own = owner[i];
  const int mc  = matchcnt[i];

  float scale = 1.0f;
  if (own < 0 && mc > 0) {
    for (int d = tid; d < D_SZ; d += 256) acc[d] = 0.f;   // each thread owns d%256==tid
    int c = 0;
    for (int b = 0; b < B_SZ; ++b) {
      if (!novel_ws[b] && bestidx_ws[b] == i) {
        ++c;
        const float* pat = patterns + (size_t)b * D_SZ;
        for (int d = tid; d < D_SZ; d += 256)
          acc[d] = MOM * acc[d] + (1.0f - MOM) * pat[d];
      }
    }
    scale = powf(MOM, (float)c);
  }
  __syncthreads();

  float* mo = mem_out + (size_t)i * Q_SZ * D_SZ;
  const float* mi = qm + (size_t)i * Q_SZ * D_SZ;
  if (own >= 0) {
    const float* pat = patterns + (size_t)own * D_SZ;
    for (int q = 0; q < Q_SZ; ++q) {
      const float s2 = 0.5f + 0.5f * rand_scale[own * Q_SZ + q];
      for (int d = tid; d < D_SZ; d += 256) mo[(size_t)q * D_SZ + d] = pat[d] * s2;
    }
    if (tid < Q_SZ) ph_out[(size_t)i * Q_SZ + tid] = rand_phase[own * Q_SZ + tid] * TWO_PI;
  } else if (mc > 0) {
    for (int q = 0; q < Q_SZ; ++q)
      for (int d = tid; d < D_SZ; d += 256)
        mo[(size_t)q * D_SZ + d] = scale * mi[(size_t)q * D_SZ + d] + acc[d];
    if (tid < Q_SZ) ph_out[(size_t)i * Q_SZ + tid] = qph[(size_t)i * Q_SZ + tid];
  } else {
    for (int q = 0; q < Q_SZ; ++q)
      for (int d = tid; d < D_SZ; d += 256)
        mo[(size_t)q * D_SZ + d] = mi[(size_t)q * D_SZ + d];
    if (tid < Q_SZ) ph_out[(size_t)i * Q_SZ + tid] = qph[(size_t)i * Q_SZ + tid];
  }
}

// ---------------- usage / last_access bookkeeping (add-once-per-unique-index) ----------------
__global__ __launch_bounds__(256)
void k_usage(const float* __restrict__ usage, const int* __restrict__ last_access,
             const int* __restrict__ time_step, const int* __restrict__ mids,
             float* __restrict__ usage_out, float* __restrict__ last_out) {
  int i = blockIdx.x * blockDim.x + threadIdx.x;
  if (i >= M_SZ) return;
  int t = 0;
  for (int b = 0; b < B_SZ; ++b) if (mids[b] == i) { t = 1; break; }
  usage_out[i] = 0.99f * usage[i] + (t ? 1.0f : 0.0f);
  last_out[i]  = t ? (float)time_step[0] : (float)last_access[i];
}

extern "C" void kernel_launch(void* const* d_in, const int* in_sizes, int n_in,
                              void* d_out, int out_size, void* d_ws, size_t ws_size,
                              hipStream_t stream) {
  const float* patterns   = (const float*)d_in[0];
  const float* qmem       = (const float*)d_in[1];
  const float* qph        = (const float*)d_in[2];
  const float* usage      = (const float*)d_in[3];
  const float* rand_scale = (const float*)d_in[4];
  const float* rand_phase = (const float*)d_in[5];
  const float* W1         = (const float*)d_in[6];
  const float* b1         = (const float*)d_in[7];
  const float* W2         = (const float*)d_in[8];
  const float* b2         = (const float*)d_in[9];
  const int*   last_acc   = (const int*)d_in[10];
  const int*   time_step  = (const int*)d_in[11];

  float* out = (float*)d_out;
  // output layout (flat, return order)
  float* o_meaning = out;                          // 1024
  float* o_isnovel = out + 1024;                   // 1024
  float* o_conf    = out + 2048;                   // 1024
  float* o_sims    = out + 3072;                   // 1024*8192
  float* o_thr     = out + 3072 + (size_t)B_SZ * M_SZ;          // 1
  float* o_mem     = o_thr + 1;                                  // 8192*4*2048
  float* o_ph      = o_mem + (size_t)M_SZ * Q_SZ * D_SZ;         // 8192*4
  float* o_usage   = o_ph + (size_t)M_SZ * Q_SZ;                 // 8192
  float* o_last    = o_usage + M_SZ;                             // 8192

  // workspace bump allocator (256B aligned)
  char* ws = (char*)d_ws;
  size_t off = 0;
  auto alloc = [&](size_t bytes) { void* p = ws + off; off = (off + bytes + 255) & ~(size_t)255; return p; };
  float* mn       = (float*)alloc((size_t)M_SZ * D_SZ * 4);
  float* pn       = (float*)alloc((size_t)B_SZ * D_SZ * 4);
  float* rowsum   = (float*)alloc(B_SZ * 4);
  float* rowsq    = (float*)alloc(B_SZ * 4);
  float* bestsim  = (float*)alloc(B_SZ * 4);
  int*   bestidx  = (int*)  alloc(B_SZ * 4);
  float* thr_ws   = (float*)alloc(256);
  float* imp      = (float*)alloc(M_SZ * 4);
  int*   pool     = (int*)  alloc(B_SZ * 4);
  int*   novel_ws = (int*)  alloc(B_SZ * 4);
  int*   mids_ws  = (int*)  alloc(B_SZ * 4);
  int*   owner    = (int*)  alloc(M_SZ * 4);
  int*   matchcnt = (int*)  alloc(M_SZ * 4);

  // 1) normalized mean bank + normalized patterns
  k_mean_norm<<<M_SZ, 256, 0, stream>>>(qmem, mn, Q_SZ);
  k_mean_norm<<<B_SZ, 256, 0, stream>>>(patterns, pn, 1);
  // 2) sims GEMM (WMMA f32)
  k_gemm_wmma<<<dim3(M_SZ / 256, B_SZ / 16), 128, 0, stream>>>(pn, mn, o_sims);
  // 3) per-row stats
  k_rowstats<<<B_SZ, 256, 0, stream>>>(o_sims, rowsum, rowsq, bestsim, bestidx);
  // 4) threshold MLP
  k_thresh<<<1, 256, 0, stream>>>(rowsum, rowsq, bestsim, usage, W1, b1, W2, b2, thr_ws, o_thr);
  // 5) importance + aux init
  k_imp<<<M_SZ / 256, 256, 0, stream>>>(usage, last_acc, time_step, imp, owner, matchcnt);
  // 6) stable top-k eviction pool
  k_pool<<<M_SZ / 256, 256, 0, stream>>>(imp, pool);
  // 7) novelty / rank / meaning ids
  k_novel<<<1, 1024, 0, stream>>>(bestsim, bestidx, thr_ws, pool, novel_ws, mids_ws,
                                  owner, matchcnt, o_meaning, o_isnovel, o_conf);
  // 8) memory + phase update
  k_mem<<<M_SZ, 256, 0, stream>>>(qmem, qph, patterns, rand_scale, rand_phase,
                                  novel_ws, bestidx, owner, matchcnt, o_mem, o_ph);
  // 9) usage / last_access
  k_usage<<<M_SZ / 256, 256, 0, stream>>>(usage, last_acc, time_step, mids_ws, o_usage, o_last);
}